// Renderer_36189394436976
// MI455X (gfx1250) — compile-verified
//
#include <hip/hip_runtime.h>
#include <math.h>

typedef __attribute__((ext_vector_type(2))) float v2f;
typedef __attribute__((ext_vector_type(8))) float v8f;

#define BT 16
#define NPTS 25000
#define DDIM 195          // 3 + 3*64
#define TOTAL (BT * NPTS) // 400000
#define CHUNKS (TOTAL / 32) // 12500, exact
#define NSTEP 49          // K padded 195 -> 196 = 49 * 4
#define LDSTRIDE 17       // 32x17 f32 tile per wave; 17 coprime w/ 64 banks

__launch_bounds__(256)
__global__ void renderer_wmma_kernel(
    const float* __restrict__ pts,  const float* __restrict__ feat,
    const float* __restrict__ Wxyz, const float* __restrict__ bxyz,
    const float* __restrict__ Wrot, const float* __restrict__ brot,
    const float* __restrict__ Wscl, const float* __restrict__ bscl,
    const float* __restrict__ Wop,  const float* __restrict__ bop,
    const float* __restrict__ Wshs, const float* __restrict__ bshs,
    const float* __restrict__ Kmat, const float* __restrict__ Emat,
    float* __restrict__ out, int nwaves)
{
  __shared__ float fpLDS[BT * 16];              // full_proj per batch, [b][i][k]
  __shared__ float cLDS[8 * 32 * LDSTRIDE];     // per-wave 32x16 accum transpose

  const int tid   = threadIdx.x;
  const int lane  = tid & 31;
  const int wslot = tid >> 5;
  const int n     = lane & 15;      // B/C column this lane owns
  const int lh    = lane >> 4;      // lane half
  const int kbase = lh * 2;         // K offset within each 4-wide K block
  const int wid   = (blockIdx.x * blockDim.x + tid) >> 5;

  // ---- per-block: build full_proj[b] = E[b]^T (4x4-sparse P[b])^T into LDS ----
  {
    int b = tid >> 4, idx = tid & 15, i = idx >> 2, k = idx & 3;
    float fx = Kmat[b * 9 + 0], fy = Kmat[b * 9 + 4];
    float cx = Kmat[b * 9 + 2], cy = Kmat[b * 9 + 5];
    float p0 = 0.f, p1 = 0.f, p2 = 0.f, p3 = 0.f; // P[k][0..3]
    if (k == 0)      { p0 = 2.f * fx / 512.f; p2 = 2.f * cx / 512.f - 1.f; }
    else if (k == 1) { p1 = 2.f * fy / 512.f; p2 = 2.f * cy / 512.f - 1.f; }
    else if (k == 2) { p2 = 100.f / (100.f - 0.01f);
                       p3 = -(100.f * 0.01f) / (100.f - 0.01f); }
    else             { p2 = 1.f; }
    const float* Eb = Emat + b * 16;
    // full_proj[i][k] = sum_j E[j][i] * P[k][j]
    fpLDS[b * 16 + i * 4 + k] =
        Eb[0 * 4 + i] * p0 + Eb[1 * 4 + i] * p1 +
        Eb[2 * 4 + i] * p2 + Eb[3 * 4 + i] * p3;
  }
  __syncthreads();

  // ---- preload B fragments: combined 196x16 weight matrix (col n per lane) ----
  const float* wp = Wxyz; int ncols = 3, col = 0; float wscale = 1.f;
  if (n < 3)        { wp = Wxyz; ncols = 3; col = n; }
  else if (n < 7)   { wp = Wrot; ncols = 4; col = n - 3; }
  else if (n < 10)  { wp = Wscl; ncols = 3; col = n - 7; }
  else if (n == 10) { wp = Wop;  ncols = 1; col = 0; }
  else if (n < 14)  { wp = Wshs; ncols = 3; col = n - 11; }
  else              { wp = Wxyz; ncols = 3; col = 0; wscale = 0.f; } // pad cols

  v2f bfrag[NSTEP];
  #pragma unroll
  for (int s = 0; s < NSTEP; ++s) {
    int k0 = 4 * s + kbase;
    int ka = (k0     < DDIM) ? k0     : (DDIM - 1); // clamp: keep loads in-bounds
    int kb = (k0 + 1 < DDIM) ? k0 + 1 : (DDIM - 1);
    float bx = wp[ka * ncols + col];
    float by = wp[kb * ncols + col];
    if (k0     >= DDIM) bx = 0.f; // K pad row (k == 195)
    if (k0 + 1 >= DDIM) by = 0.f;
    v2f bf; bf.x = bx * wscale; bf.y = by * wscale;
    bfrag[s] = bf;
  }

  // ---- grid-stride over chunks of 32 points ----
  for (int chunk = wid; chunk < CHUNKS; chunk += nwaves) {
    const int gbase = chunk * 32;
    v8f acc[2];

    #pragma unroll
    for (int t = 0; t < 2; ++t) {
      const int p = gbase + t * 16 + n;              // A row (point) this lane feeds
      const float* rp = pts  + p * 3;
      const float* rf = feat + p * 192;
      v8f c = {};

      // s = 0: K = 0..3 straddles points/features
      {
        v2f a;
        if (lh == 0) { a.x = rp[0]; a.y = rp[1]; }   // K=0,1
        else         { a.x = rp[2]; a.y = rf[0]; }   // K=2,3
        c = __builtin_amdgcn_wmma_f32_16x16x4_f32(false, a, false, bfrag[0],
                                                  (short)0, c, false, false);
      }
      // s = 1..47: pure feature columns
      #pragma unroll
      for (int s = 1; s < 48; ++s) {
        int kf = 4 * s + kbase - 3;
        v2f a; a.x = rf[kf]; a.y = rf[kf + 1];
        c = __builtin_amdgcn_wmma_f32_16x16x4_f32(false, a, false, bfrag[s],
                                                  (short)0, c, false, false);
      }
      // s = 48: K = 192..194 valid, K = 195 is zero pad
      {
        v2f a;
        a.x = rf[189 + kbase];                       // K = 192 + 2*lh
        a.y = (lh == 0) ? rf[190] : 0.f;             // K = 193 / 195(pad)
        c = __builtin_amdgcn_wmma_f32_16x16x4_f32(false, a, false, bfrag[48],
                                                  (short)0, c, false, false);
      }
      acc[t] = c;
    }

    // ---- transpose 32x14 accum through LDS so each lane owns one point ----
    float* myC = &cLDS[wslot * 32 * LDSTRIDE];
    #pragma unroll
    for (int j = 0; j < 8; ++j) {
      myC[(8 * lh + j) * LDSTRIDE + n]      = acc[0][j]; // rows 0..15
      myC[(16 + 8 * lh + j) * LDSTRIDE + n] = acc[1][j]; // rows 16..31
    }
    asm volatile("s_wait_dscnt 0x0" ::: "memory");       // wave-local LDS ordering

    float v[14];
    #pragma unroll
    for (int cidx = 0; cidx < 14; ++cidx)
      v[cidx] = myC[lane * LDSTRIDE + cidx];
    asm volatile("" ::: "memory");

    // ---- per-point post-processing ----
    const int g = gbase + lane;
    const float px = pts[g * 3 + 0], py = pts[g * 3 + 1], pz = pts[g * 3 + 2];
    const float X = px + v[0] + bxyz[0];
    const float Y = py + v[1] + bxyz[1];
    const float Z = pz + v[2] + bxyz[2];

    float r0 = v[3] + brot[0], r1 = v[4] + brot[1];
    float r2 = v[5] + brot[2], r3 = v[6] + brot[3];
    const float rn = 1.f / (sqrtf(r0 * r0 + r1 * r1 + r2 * r2 + r3 * r3) + 1e-8f);
    r0 *= rn; r1 *= rn; r2 *= rn; r3 *= rn;

    const float s0 = fminf(expf(v[7] + bscl[0] - 3.9f), 0.1f);
    const float s1 = fminf(expf(v[8] + bscl[1] - 3.9f), 0.1f);
    const float s2 = fminf(expf(v[9] + bscl[2] - 3.9f), 0.1f);

    const float opac = 1.f / (1.f + expf(-(v[10] + bop[0])));

    const float c0 = fminf(fmaxf(v[11] + bshs[0], 0.f), 1.f);
    const float c1 = fminf(fmaxf(v[12] + bshs[1], 0.f), 1.f);
    const float c2 = fminf(fmaxf(v[13] + bshs[2], 0.f), 1.f);

    const int b = g / NPTS;
    const float* fp = &fpLDS[b * 16];
    const float clip0 = X * fp[0] + Y * fp[4] + Z * fp[8]  + fp[12];
    const float clip1 = X * fp[1] + Y * fp[5] + Z * fp[9]  + fp[13];
    const float clip2 = X * fp[2] + Y * fp[6] + Z * fp[10] + fp[14];
    const float clip3 = X * fp[3] + Y * fp[7] + Z * fp[11] + fp[15];
    const float invw = 1.f / (clip3 + 1e-7f);

    float* o = out + g * 14;
    o[0]  = clip0 * invw; o[1]  = clip1 * invw; o[2] = clip2 * invw;
    o[3]  = s0;  o[4]  = s1;  o[5] = s2;
    o[6]  = opac;
    o[7]  = c0;  o[8]  = c1;  o[9] = c2;
    o[10] = r0;  o[11] = r1;  o[12] = r2;  o[13] = r3;
  }
}

extern "C" void kernel_launch(void* const* d_in, const int* in_sizes, int n_in,
                              void* d_out, int out_size, void* d_ws, size_t ws_size,
                              hipStream_t stream) {
  (void)in_sizes; (void)n_in; (void)out_size; (void)d_ws; (void)ws_size;
  const float* pts  = (const float*)d_in[0];
  const float* feat = (const float*)d_in[1];
  const float* Wxyz = (const float*)d_in[2];
  const float* bxyz = (const float*)d_in[3];
  const float* Wrot = (const float*)d_in[4];
  const float* brot = (const float*)d_in[5];
  const float* Wscl = (const float*)d_in[6];
  const float* bscl = (const float*)d_in[7];
  const float* Wop  = (const float*)d_in[8];
  const float* bop  = (const float*)d_in[9];
  const float* Wshs = (const float*)d_in[10];
  const float* bshs = (const float*)d_in[11];
  const float* Kmat = (const float*)d_in[12];
  const float* Emat = (const float*)d_in[13];
  float* out = (float*)d_out;

  const int threads = 256;                 // 8 wave32s per block
  const int blocks  = 512;                 // 4096 waves, ~3 chunks each
  const int nwaves  = blocks * threads / 32;
  renderer_wmma_kernel<<<blocks, threads, 0, stream>>>(
      pts, feat, Wxyz, bxyz, Wrot, brot, Wscl, bscl, Wop, bop, Wshs, bshs,
      Kmat, Emat, out, nwaves);
}